// PrimalBlock_12060268167684
// MI455X (gfx1250) — compile-verified
//
#include <hip/hip_runtime.h>

// Problem constants (from reference setup_inputs)
#define BATCH 4
#define CH    3
#define H     256
#define W     256
#define HW    (H * W)
#define NEL   (BATCH * CH * HW)     // 786432
#define NEL4  (NEL / 4)             // 196608
#define NB    256                   // blocks (== number of 32x32 tiles: 4 * 8 * 8)
#define NT    256                   // threads per block (8 wave32)
#define TOTW  (NB * NT)             // 65536
#define TS    32                    // output tile
#define DT    36                    // d tile with halo 2
#define ET    34                    // e tile with halo 1
#define TOLV  1e-6f
#define NITER 101

typedef __attribute__((ext_vector_type(2))) float v2f;
typedef __attribute__((ext_vector_type(8))) float v8f;

struct SharedMem {
    float d[DT][DT + 1];            // Sum_c mask*p (or y), zero-padded outside image
    float e[CH][ET][ET + 1];        // P(d), zero at out-of-image positions
    float red[NT];
    float kk[32];                   // 3x3x3 kernel
    float bcast;
};

// ---------------- grid-wide barrier (monotone generation, all-thread fences) --
__device__ __forceinline__ void grid_barrier(unsigned* cnt, unsigned* rel, unsigned gen) {
    __threadfence();                 // make my stores device-visible
    __syncthreads();
    if (threadIdx.x == 0) {
        unsigned t = __hip_atomic_fetch_add(cnt, 1u, __ATOMIC_ACQ_REL,
                                            __HIP_MEMORY_SCOPE_AGENT) + 1u;
        if (t == gen * (unsigned)NB) {
            __hip_atomic_store(rel, gen, __ATOMIC_RELEASE, __HIP_MEMORY_SCOPE_AGENT);
        } else {
            while (__hip_atomic_load(rel, __ATOMIC_ACQUIRE, __HIP_MEMORY_SCOPE_AGENT) < gen) {
                __builtin_amdgcn_s_sleep(2);
            }
        }
    }
    __syncthreads();
    __threadfence();                 // every CU invalidates its L0 before reading
}

// ---------------- deterministic block reduce into per-block slot --------------
__device__ __forceinline__ void block_reduce_to(float part, float* slot, SharedMem& sm) {
    sm.red[threadIdx.x] = part;
    __syncthreads();
    #pragma unroll
    for (int s = NT / 2; s > 0; s >>= 1) {
        if ((int)threadIdx.x < s) sm.red[threadIdx.x] += sm.red[threadIdx.x + s];
        __syncthreads();
    }
    if (threadIdx.x == 0) slot[blockIdx.x] = sm.red[0];
    __syncthreads();
}

// ---------------- deterministic reduce-all of the 256 slots -------------------
// Wave 0 reduces; half-wave pair-sum done with V_WMMA_F32_16X16X4_F32:
//   A = ones(16x4), B vgpr0 = lane partials, B vgpr1 = 0
//   => D[0][n] = s_n + s_{n+16}, identical for all rows, so D.s0 holds t_{lane&15}.
__device__ __forceinline__ float reduce_all_slots(const float* slots, SharedMem& sm) {
    if (threadIdx.x < 32) {
        float s = 0.0f;
        #pragma unroll
        for (int i = 0; i < 8; ++i) s += slots[threadIdx.x * 8 + i];
#if __has_builtin(__builtin_amdgcn_wmma_f32_16x16x4_f32)
        v2f a = {1.0f, 1.0f};
        v2f b = {s, 0.0f};
        v8f c = {};
        c = __builtin_amdgcn_wmma_f32_16x16x4_f32(false, a, false, b,
                                                  (short)0, c, false, false);
        float t = c[0];                       // t_{lane & 15} in every lane
        // xor-swizzle tree over masks 1,2,4,8 (value independent of lane bit 4)
        t += __int_as_float(__builtin_amdgcn_ds_swizzle(__float_as_int(t), (1 << 10) | 0x1F));
        t += __int_as_float(__builtin_amdgcn_ds_swizzle(__float_as_int(t), (2 << 10) | 0x1F));
        t += __int_as_float(__builtin_amdgcn_ds_swizzle(__float_as_int(t), (4 << 10) | 0x1F));
        t += __int_as_float(__builtin_amdgcn_ds_swizzle(__float_as_int(t), (8 << 10) | 0x1F));
#else
        float t = s;
        t += __shfl_xor(t, 16);
        t += __shfl_xor(t, 8);
        t += __shfl_xor(t, 4);
        t += __shfl_xor(t, 2);
        t += __shfl_xor(t, 1);
#endif
        if (threadIdx.x == 0) sm.bcast = t;
    }
    __syncthreads();
    float v = sm.bcast;
    __syncthreads();
    return v;
}

// ---------------- fused p-update -> mask-reduce -> P -> PT -> outputs ---------
// MODE 0: d from y; per-pixel writes x = b = mask*g + rho*(z-beta).
// MODE 1: p_cur = rsrc + bt*p_prev (fused); d = Sum_c mask*p_cur;
//         writes p_cur and Ap = rho*p_cur + mask*g; returns Sum p_cur*Ap partial.
template <int MODE>
__device__ float op_tile(SharedMem& sm,
                         const float* __restrict__ mask, const float* __restrict__ y,
                         const float* __restrict__ z, const float* __restrict__ bet,
                         float rho, float* __restrict__ x,
                         const float* __restrict__ rsrc,
                         const float* __restrict__ pprev, float bt,
                         float* __restrict__ pcur, float* __restrict__ Ap) {
    const int tid = threadIdx.x;
    const int bid = blockIdx.x;
    const int bimg = bid >> 6;               // image index (4 images, 64 tiles each)
    const int t = bid & 63;
    const int oy0 = (t >> 3) * TS;
    const int ox0 = (t & 7) * TS;
    const int ibase = bimg * CH * HW;

    // Stage 0: d tile with halo 2 (zero outside image == conv zero padding)
    for (int idx = tid; idx < DT * DT; idx += NT) {
        const int i = idx / DT, j = idx % DT;
        const int gy = oy0 - 2 + i, gx = ox0 - 2 + j;
        float v = 0.0f;
        if (gy >= 0 && gy < H && gx >= 0 && gx < W) {
            if (MODE == 0) {
                v = y[bimg * HW + gy * W + gx];
            } else {
                const int a = ibase + gy * W + gx;
                #pragma unroll
                for (int c = 0; c < CH; ++c) {
                    const int ac = a + c * HW;
                    const float pc = fmaf(bt, pprev[ac], rsrc[ac]);
                    v = fmaf(mask[ac], pc, v);
                }
            }
        }
        sm.d[i][j] = v;
    }
    __syncthreads();

    // Stage 1: e = P(d), halo 1, ZERO at out-of-image positions (PT's padding)
    for (int idx = tid; idx < CH * ET * ET; idx += NT) {
        const int c = idx / (ET * ET);
        const int rem = idx % (ET * ET);
        const int ey = rem / ET, ex = rem % ET;
        const int gy = oy0 - 1 + ey, gx = ox0 - 1 + ex;
        float v = 0.0f;
        if (gy >= 0 && gy < H && gx >= 0 && gx < W) {
            const float* kc = &sm.kk[c * 9];
            #pragma unroll
            for (int u = 0; u < 3; ++u)
                #pragma unroll
                for (int w2 = 0; w2 < 3; ++w2)
                    v = fmaf(kc[u * 3 + w2], sm.d[ey + u][ex + w2], v);
        }
        sm.e[c][ey][ex] = v;
    }
    __syncthreads();

    // Stage 2: g = PT(e) (flipped kernel, summed over channels) + per-channel out
    float part = 0.0f;
    for (int idx = tid; idx < TS * TS; idx += NT) {
        const int oy = idx / TS, ox = idx % TS;
        float g = 0.0f;
        #pragma unroll
        for (int c = 0; c < CH; ++c)
            #pragma unroll
            for (int u = 0; u < 3; ++u)
                #pragma unroll
                for (int w2 = 0; w2 < 3; ++w2)
                    g = fmaf(sm.kk[c * 9 + (2 - u) * 3 + (2 - w2)],
                             sm.e[c][oy + u][ox + w2], g);
        const int pix = (oy0 + oy) * W + (ox0 + ox);
        #pragma unroll
        for (int c = 0; c < CH; ++c) {
            const int a = ibase + c * HW + pix;
            if (MODE == 0) {
                x[a] = fmaf(mask[a], g, rho * (z[a] - bet[a]));
            } else {
                const float pcv = fmaf(bt, pprev[a], rsrc[a]);
                pcur[a] = pcv;
                const float ap = fmaf(mask[a], g, rho * pcv);
                Ap[a] = ap;
                part = fmaf(pcv, ap, part);
            }
        }
    }
    return part;
}

// ---------------- init kernel: zero barrier state each launch -----------------
__global__ void cg_init(unsigned* cnt, unsigned* rel) {
    if (threadIdx.x == 0) { *cnt = 0u; *rel = 0u; }
}

// ---------------- persistent CG kernel ---------------------------------------
__global__ void __launch_bounds__(NT, 1)
cg_main(const float* __restrict__ mask, const float* __restrict__ y,
        const float* __restrict__ z, const float* __restrict__ bet,
        const float* __restrict__ rho_p, const float* __restrict__ ker,
        float* __restrict__ x, float* __restrict__ r,
        float* __restrict__ pbufA, float* __restrict__ pbufB,
        float* __restrict__ Ap, unsigned* cnt, unsigned* rel,
        float* slots0, float* slots1) {
    __shared__ SharedMem sm;
    const int tid = threadIdx.x;
    const int gid = blockIdx.x * NT + tid;

    if (tid < 27) sm.kk[tid] = ker[tid];
    __syncthreads();
    const float rho = rho_p[0];
    unsigned gen = 0;

    float4* const x4 = (float4*)x;
    float4* const r4 = (float4*)r;
    const float4* const ap4 = (const float4*)Ap;

    // Init A: b = mask*PT(P(y)) + rho*(z-beta); x = b
    op_tile<0>(sm, mask, y, z, bet, rho, x, x, x, 0.0f, pbufA, Ap);
    grid_barrier(cnt, rel, ++gen);

    // Init B: Ap = A(b)  (fused op with rsrc=x=b, bt=0; scratch p target pbufA)
    op_tile<1>(sm, mask, y, z, bet, rho, x, x, pbufB, 0.0f, pbufA, Ap);
    grid_barrier(cnt, rel, ++gen);

    // Init C: r = b - A(b); crit = ||r||^2   (vectorized)
    float part = 0.0f;
    for (int i = gid; i < NEL4; i += TOTW) {
        const float4 xv = x4[i];
        const float4 av = ap4[i];
        float4 rv;
        rv.x = xv.x - av.x; rv.y = xv.y - av.y;
        rv.z = xv.z - av.z; rv.w = xv.w - av.w;
        r4[i] = rv;
        part = fmaf(rv.x, rv.x, part); part = fmaf(rv.y, rv.y, part);
        part = fmaf(rv.z, rv.z, part); part = fmaf(rv.w, rv.w, part);
    }
    block_reduce_to(part, slots1, sm);
    grid_barrier(cnt, rel, ++gen);
    float crit = reduce_all_slots(slots1, sm);

    float bt = 0.0f;                 // iteration 0: p = r + 0*p_prev == r0
    float* pt = pbufA;               // current-p target
    float* pp = pbufB;               // previous p
    for (int it = 0; it < NITER; ++it) {
        if (crit < TOLV) break;      // uniform across all threads/blocks

        // Phase 1: p = r + bt*p_prev (fused); Ap = A(p); denom = <p, Ap>
        const float dpart = op_tile<1>(sm, mask, y, z, bet, rho, x, r, pp, bt, pt, Ap);
        block_reduce_to(dpart, slots0, sm);
        grid_barrier(cnt, rel, ++gen);
        const float denom = reduce_all_slots(slots0, sm);
        const float alpha = crit / denom;

        // Phase 2: x += alpha p; r -= alpha Ap; crit_n = ||r||^2   (vectorized)
        const float4* const pt4 = (const float4*)pt;
        float p2 = 0.0f;
        for (int i = gid; i < NEL4; i += TOTW) {
            const float4 pv = pt4[i];
            const float4 av = ap4[i];
            float4 xv = x4[i];
            float4 rv = r4[i];
            xv.x = fmaf(alpha, pv.x, xv.x); xv.y = fmaf(alpha, pv.y, xv.y);
            xv.z = fmaf(alpha, pv.z, xv.z); xv.w = fmaf(alpha, pv.w, xv.w);
            rv.x = fmaf(-alpha, av.x, rv.x); rv.y = fmaf(-alpha, av.y, rv.y);
            rv.z = fmaf(-alpha, av.z, rv.z); rv.w = fmaf(-alpha, av.w, rv.w);
            x4[i] = xv;
            r4[i] = rv;
            p2 = fmaf(rv.x, rv.x, p2); p2 = fmaf(rv.y, rv.y, p2);
            p2 = fmaf(rv.z, rv.z, p2); p2 = fmaf(rv.w, rv.w, p2);
        }
        block_reduce_to(p2, slots1, sm);
        grid_barrier(cnt, rel, ++gen);
        const float critn = reduce_all_slots(slots1, sm);

        bt = critn / crit;           // beta for the fused p-update next iteration
        crit = critn;
        float* tmp = pt; pt = pp; pp = tmp;   // double-buffer swap
    }
}

extern "C" void kernel_launch(void* const* d_in, const int* in_sizes, int n_in,
                              void* d_out, int out_size, void* d_ws, size_t ws_size,
                              hipStream_t stream) {
    const float* mask = (const float*)d_in[0];
    const float* y    = (const float*)d_in[1];
    const float* z    = (const float*)d_in[2];
    const float* bet  = (const float*)d_in[3];
    const float* rho  = (const float*)d_in[4];
    const float* ker  = (const float*)d_in[5];
    float* x = (float*)d_out;

    float* wsf = (float*)d_ws;
    unsigned* cnt = (unsigned*)d_ws;          // ws[0]
    unsigned* rel = cnt + 1;                  // ws[1]
    float* slots0 = wsf + 16;                 // 256 floats
    float* slots1 = wsf + 16 + 256;           // 256 floats
    float* rbuf   = wsf + 1024;               // NEL floats (16B-aligned offsets)
    float* pbufA  = rbuf + NEL;
    float* pbufB  = pbufA + NEL;
    float* apbuf  = pbufB + NEL;

    cg_init<<<1, 32, 0, stream>>>(cnt, rel);
    cg_main<<<NB, NT, 0, stream>>>(mask, y, z, bet, rho, ker,
                                   x, rbuf, pbufA, pbufB, apbuf,
                                   cnt, rel, slots0, slots1);
}